// EnvironmentSpecificDecoder_82240033784132
// MI455X (gfx1250) — compile-verified
//
#include <hip/hip_runtime.h>
#include <math.h>

typedef float v2f __attribute__((ext_vector_type(2)));
typedef float v8f __attribute__((ext_vector_type(8)));

#define WMMA_F32(a, b, c) \
  __builtin_amdgcn_wmma_f32_16x16x4_f32(false, (a), false, (b), (short)0, (c), false, false)

constexpr int B_ = 32, T_ = 64, D_ = 128, L_ = 64, H_ = 256, H2_ = 128, NE_ = 8;

__global__ __launch_bounds__(128, 1)
void esd_fused_kernel(const float* __restrict__ zs, const float* __restrict__ zc,
                      const float* __restrict__ A,  const int*   __restrict__ regime,
                      const float* __restrict__ Wsig, const float* __restrict__ bsig,
                      const float* __restrict__ W1e,  const float* __restrict__ b1e,
                      const float* __restrict__ W2e,  const float* __restrict__ b2e,
                      const float* __restrict__ Wc,   const float* __restrict__ bc,
                      const float* __restrict__ Wo,   const float* __restrict__ bo,
                      float* __restrict__ out) {
  // 4 waves x (16 rows x 256 cols) fp32 private slabs = 64 KB
  __shared__ float slab[4 * 16 * 256];
  // block-shared weight panel: 16 K-rows x 256 cols fp32 = 16 KB
  __shared__ float panel[16 * 256];

  const int bt   = blockIdx.x;          // b*T + t
  const int b    = bt / T_;
  const int tid  = threadIdx.x;
  const int lane = tid & 31;
  const int wave = tid >> 5;
  float* sw = slab + wave * (16 * 256);

  int e = regime[b];
  if (e < 0 || e >= NE_) e = 0;

  const int col   = lane & 15;          // N index (or M index for A-fragments)
  const int khalf = (lane >> 4) * 2;    // K base: lanes 0-15 -> K{0,1}, 16-31 -> K{2,3}
  const int mrow  = (lane >> 4) * 8;    // C/D layout: lanes 16-31 hold M = v+8

  const float* zs_bt = zs + (size_t)bt * D_ * L_;
  const float* zc_bt = zc + (size_t)bt * D_ * L_;
  const float* W1    = W1e + (size_t)e * H_ * H_;

  for (int rep = 0; rep < 2; ++rep) {
    const int m0 = (wave * 2 + rep) * 16;   // this wave-tile's 16 rows of d

    // ---- Step A: z_agg(16x64) = A^T[m0:m0+16,:] @ Zs(128x64) ----
    // A (64KB) and the per-block Zs tile (32KB) are WGP$/L2 resident; direct loads.
    v8f Cz[4];
#pragma unroll
    for (int n = 0; n < 4; ++n) Cz[n] = v8f{};
    for (int j = 0; j < D_; j += 4) {
      v2f a;
      a.x = A[(j + khalf + 0) * D_ + (m0 + col)];
      a.y = A[(j + khalf + 1) * D_ + (m0 + col)];
#pragma unroll
      for (int n = 0; n < 4; ++n) {
        v2f bm;
        bm.x = zs_bt[(j + khalf + 0) * L_ + n * 16 + col];
        bm.y = zs_bt[(j + khalf + 1) * L_ + n * 16 + col];
        Cz[n] = WMMA_F32(a, bm, Cz[n]);
      }
    }
    // stage z_agg into slab, stride 64 (D-layout -> row major)
#pragma unroll
    for (int n = 0; n < 4; ++n)
#pragma unroll
      for (int v = 0; v < 8; ++v)
        sw[(v + mrow) * 64 + n * 16 + col] = Cz[n][v];

    // ---- Step B: h(16x256) = z_agg(16x64) @ Wsig(64x256) + bsig ----
    v8f Ch[16];
#pragma unroll
    for (int nt = 0; nt < 16; ++nt) {
      float bs = bsig[nt * 16 + col];
#pragma unroll
      for (int v = 0; v < 8; ++v) Ch[nt][v] = bs;
    }
    for (int kp = 0; kp < 4; ++kp) {            // K-panels of 16 rows of Wsig
      __syncthreads();
      {
        const float4* src = reinterpret_cast<const float4*>(Wsig + kp * 16 * H_);
        float4* dst = reinterpret_cast<float4*>(panel);
#pragma unroll
        for (int i = 0; i < 8; ++i) dst[tid + i * 128] = src[tid + i * 128];
      }
      __syncthreads();
      for (int hl = 0; hl < 16; hl += 4) {
        const int l = kp * 16 + hl;
        v2f a;
        a.x = sw[col * 64 + l + khalf + 0];
        a.y = sw[col * 64 + l + khalf + 1];
#pragma unroll
        for (int nt = 0; nt < 16; ++nt) {
          v2f bm;
          bm.x = panel[(hl + khalf + 0) * H_ + nt * 16 + col];
          bm.y = panel[(hl + khalf + 1) * H_ + nt * 16 + col];
          Ch[nt] = WMMA_F32(a, bm, Ch[nt]);
        }
      }
    }
    // stage h into slab, stride 256
#pragma unroll
    for (int nt = 0; nt < 16; ++nt)
#pragma unroll
      for (int v = 0; v < 8; ++v)
        sw[(v + mrow) * 256 + nt * 16 + col] = Ch[nt][v];

    // ---- Step C: h1(16x256) = relu(h @ W1[e](256x256) + b1[e]) ----
    v8f C1[16];
#pragma unroll
    for (int nt = 0; nt < 16; ++nt) {
      float b1v = b1e[e * H_ + nt * 16 + col];
#pragma unroll
      for (int v = 0; v < 8; ++v) C1[nt][v] = b1v;
    }
    for (int kp = 0; kp < 16; ++kp) {           // K-panels of 16 rows of W1[e]
      __syncthreads();
      {
        const float4* src = reinterpret_cast<const float4*>(W1 + kp * 16 * H_);
        float4* dst = reinterpret_cast<float4*>(panel);
#pragma unroll
        for (int i = 0; i < 8; ++i) dst[tid + i * 128] = src[tid + i * 128];
      }
      __syncthreads();
      for (int hl = 0; hl < 16; hl += 4) {
        const int hh = kp * 16 + hl;
        v2f a;
        a.x = sw[col * 256 + hh + khalf + 0];
        a.y = sw[col * 256 + hh + khalf + 1];
#pragma unroll
        for (int nt = 0; nt < 16; ++nt) {
          v2f bm;
          bm.x = panel[(hl + khalf + 0) * H_ + nt * 16 + col];
          bm.y = panel[(hl + khalf + 1) * H_ + nt * 16 + col];
          C1[nt] = WMMA_F32(a, bm, C1[nt]);
        }
      }
    }
    // relu + stage h1 into slab, stride 256
#pragma unroll
    for (int nt = 0; nt < 16; ++nt)
#pragma unroll
      for (int v = 0; v < 8; ++v)
        sw[(v + mrow) * 256 + nt * 16 + col] = fmaxf(C1[nt][v], 0.f);

    // ---- Step D: out(16x2) = h1 @ W2[e](256x2) + b2 (zero-padded 16-col tile) ----
    // W2[e] is 2 KB; direct loads are trivial traffic.
    v8f Co;
    {
      float b2v = (col < 2) ? b2e[e * 2 + col] : 0.f;
#pragma unroll
      for (int v = 0; v < 8; ++v) Co[v] = b2v;
    }
    const int cc = (col < 2) ? col : 0;   // clamp to stay in bounds
    for (int hh = 0; hh < H_; hh += 4) {
      v2f a;
      a.x = sw[col * 256 + hh + khalf + 0];
      a.y = sw[col * 256 + hh + khalf + 1];
      v2f bm;
      float w0 = W2e[(e * H_ + hh + khalf + 0) * 2 + cc];
      float w1 = W2e[(e * H_ + hh + khalf + 1) * 2 + cc];
      bm.x = (col < 2) ? w0 : 0.f;
      bm.y = (col < 2) ? w1 : 0.f;
      Co = WMMA_F32(a, bm, Co);
    }
    // stage (mu_raw, sigma_raw) rows to slab front, stride 16
#pragma unroll
    for (int v = 0; v < 8; ++v)
      sw[(v + mrow) * 16 + col] = Co[v];

    // ---- Step E: corrupt path: relu(Zc(16x64)@Wc(64x128)+bc) @ Wo + bo ----
    v8f Cc[8];
#pragma unroll
    for (int nt = 0; nt < 8; ++nt) {
      float bcv = bc[nt * 16 + col];
#pragma unroll
      for (int v = 0; v < 8; ++v) Cc[nt][v] = bcv;
    }
    for (int kp = 0; kp < 4; ++kp) {            // K-panels of 16 rows of Wc (16x128 = 8KB)
      __syncthreads();
      {
        const float4* src = reinterpret_cast<const float4*>(Wc + kp * 16 * H2_);
        float4* dst = reinterpret_cast<float4*>(panel);
#pragma unroll
        for (int i = 0; i < 4; ++i) dst[tid + i * 128] = src[tid + i * 128];
      }
      __syncthreads();
      for (int ll = 0; ll < 16; ll += 4) {
        const int l = kp * 16 + ll;
        v2f a;
        a.x = zc_bt[(m0 + col) * L_ + l + khalf + 0];
        a.y = zc_bt[(m0 + col) * L_ + l + khalf + 1];
#pragma unroll
        for (int nt = 0; nt < 8; ++nt) {
          v2f bm;
          bm.x = panel[(ll + khalf + 0) * H2_ + nt * 16 + col];
          bm.y = panel[(ll + khalf + 1) * H2_ + nt * 16 + col];
          Cc[nt] = WMMA_F32(a, bm, Cc[nt]);
        }
      }
    }
    // contrib[M] = sum over cols of relu(Cc) * Wo[col]; reduce across 16-lane halves
    float part[8];
#pragma unroll
    for (int v = 0; v < 8; ++v) part[v] = 0.f;
#pragma unroll
    for (int nt = 0; nt < 8; ++nt) {
      float wv = Wo[nt * 16 + col];
#pragma unroll
      for (int v = 0; v < 8; ++v)
        part[v] += fmaxf(Cc[nt][v], 0.f) * wv;
    }
#pragma unroll
    for (int v = 0; v < 8; ++v) {
      part[v] += __shfl_xor(part[v], 1, 32);
      part[v] += __shfl_xor(part[v], 2, 32);
      part[v] += __shfl_xor(part[v], 4, 32);
      part[v] += __shfl_xor(part[v], 8, 32);
    }
    if (col == 0) {
#pragma unroll
      for (int v = 0; v < 8; ++v)
        sw[(v + mrow) * 16 + 2] = part[v];
    }

    // ---- Epilogue: mu = mu_raw + contrib + bo ; sigma = softplus(s_raw)+0.01 ----
    if (lane < 16) {
      const int row = lane;
      float mu = sw[row * 16 + 0] + sw[row * 16 + 2] + bo[0];
      float sr = sw[row * 16 + 1];
      float sp = (sr > 20.f) ? sr : log1pf(__expf(sr));
      size_t oidx = (size_t)bt * D_ + m0 + row;
      out[oidx] = mu;
      out[(size_t)B_ * T_ * D_ + oidx] = sp + 0.01f;
    }
  }
}

extern "C" void kernel_launch(void* const* d_in, const int* in_sizes, int n_in,
                              void* d_out, int out_size, void* d_ws, size_t ws_size,
                              hipStream_t stream) {
  (void)in_sizes; (void)n_in; (void)out_size; (void)d_ws; (void)ws_size;
  const float* zs    = (const float*)d_in[0];
  const float* zc    = (const float*)d_in[1];
  const float* A     = (const float*)d_in[2];
  const int*   reg   = (const int*)  d_in[3];
  const float* Wsig  = (const float*)d_in[4];
  const float* bsig  = (const float*)d_in[5];
  const float* W1e   = (const float*)d_in[6];
  const float* b1e   = (const float*)d_in[7];
  const float* W2e   = (const float*)d_in[8];
  const float* b2e   = (const float*)d_in[9];
  const float* Wc    = (const float*)d_in[10];
  const float* bc    = (const float*)d_in[11];
  const float* Wo    = (const float*)d_in[12];
  const float* bo    = (const float*)d_in[13];
  float* out = (float*)d_out;

  dim3 grid(B_ * T_);   // one block per (b, t)
  dim3 block(128);      // 4 waves, each owns 16 rows x 2 reps
  esd_fused_kernel<<<grid, block, 0, stream>>>(zs, zc, A, reg, Wsig, bsig,
                                               W1e, b1e, W2e, b2e, Wc, bc, Wo, bo, out);
}